// CropAndResize_20169166422184
// MI455X (gfx1250) — compile-verified
//
#include <hip/hip_runtime.h>
#include <cstdint>

// ROI-align (tf.crop_and_resize, bilinear, extrapolation 0) for
// img (4,512,64,64), rois (4,300,4) -> out (4,300,512,7,7) fp32.
// Bandwidth-bound: 120MB NT store stream + L2-resident gathers.

namespace {

constexpr int   kN        = 4;
constexpr int   kB        = 300;
constexpr int   kC        = 512;
constexpr int   kH        = 64;
constexpr int   kW        = 64;
constexpr int   kPool     = 7;
constexpr int   kPP       = kPool * kPool;   // 49
constexpr int   kPlane    = kH * kW;         // 4096
constexpr int   kPerBlock = kC * kPP;        // 25088
constexpr int   kThreads  = 256;             // 8 waves (wave32)
constexpr float kImgH     = 1024.0f;
constexpr float kImgW     = 1024.0f;

struct Desc {            // 32 B per sample point, lives in LDS
  int   off_tl, off_tr, off_bl, off_br;   // element offsets within a 64x64 plane
  float lx, ly, valmul;
  int   pad;
};

__global__ __launch_bounds__(kThreads)
void roi_align_kernel(const float* __restrict__ img,
                      const float* __restrict__ rois,
                      float* __restrict__ out)
{
  __shared__ float sroi[4];
  __shared__ Desc  sdesc[kPP];

#if __has_builtin(__builtin_amdgcn_s_wait_tensorcnt)
  __builtin_amdgcn_s_wait_tensorcnt(0);    // CDNA5 TENSORcnt wait (no-op here)
#endif

  const int blk = blockIdx.x;              // n*kB + b
  const int n   = blk / kB;
  const int tid = threadIdx.x;

  // ---- Stage the 4 ROI floats via CDNA5 async global->LDS DMA (ASYNCcnt) ----
  if (tid == 0) {
    // Generic (flat) LDS pointers carry the LDS byte offset in the low 32 bits.
    unsigned lds_off = (unsigned)(uint64_t)(&sroi[0]);
    uint64_t gaddr   = (uint64_t)(const void*)(rois + (size_t)blk * 4);
    asm volatile("global_load_async_to_lds_b128 %0, %1, off"
                 :: "v"(lds_off), "v"(gaddr)
                 : "memory");
    asm volatile("s_wait_asynccnt 0x0" ::: "memory");
  }
  __syncthreads();

  // ---- Build the 49 sample descriptors once per block (threads 0..48) ----
  if (tid < kPP) {
    const int py = tid / kPool;
    const int px = tid - py * kPool;

    const float y1 = sroi[0] / (kImgH - 1.0f);
    const float x1 = sroi[1] / (kImgW - 1.0f);
    const float y2 = sroi[2] / (kImgH - 1.0f);
    const float x2 = sroi[3] / (kImgW - 1.0f);

    const float gy = (float)py / (float)(kPool - 1);
    const float gx = (float)px / (float)(kPool - 1);

    const float in_y = (y1 + (y2 - y1) * gy) * (float)(kH - 1);
    const float in_x = (x1 + (x2 - x1) * gx) * (float)(kW - 1);

    const bool vy = (in_y >= 0.0f) && (in_y <= (float)(kH - 1));
    const bool vx = (in_x >= 0.0f) && (in_x <= (float)(kW - 1));

    const float y0f = floorf(in_y);
    const float x0f = floorf(in_x);
    const int y0 = (int)fminf(fmaxf(y0f, 0.0f), (float)(kH - 1));
    const int x0 = (int)fminf(fmaxf(x0f, 0.0f), (float)(kW - 1));
    const int yb = min(y0 + 1, kH - 1);
    const int xb = min(x0 + 1, kW - 1);

    Desc d;
    d.off_tl = y0 * kW + x0;
    d.off_tr = y0 * kW + xb;
    d.off_bl = yb * kW + x0;
    d.off_br = yb * kW + xb;
    d.lx     = in_x - x0f;                 // uses unclipped floor, as reference
    d.ly     = in_y - y0f;
    d.valmul = (vy && vx) ? 1.0f : 0.0f;
    d.pad    = 0;
    sdesc[tid] = d;
  }
  __syncthreads();

  // ---- Main gather+lerp loop: coalesced along (c*49 + p) ----
  const float* __restrict__ planeN = img + (size_t)n * kC * kPlane;
  float* __restrict__ outB         = out + (size_t)blk * kPerBlock;

  for (int e = tid; e < kPerBlock; e += kThreads) {
    const int c = e / kPP;                 // mulhi by compiler
    const int p = e - c * kPP;
    const Desc d = sdesc[p];               // ds_load_b128 x2

    const int base = c * kPlane;           // 32-bit voffset + SGPR base (GVS form)
    const float tl = planeN[base + d.off_tl];
    const float tr = planeN[base + d.off_tr];
    const float bl = planeN[base + d.off_bl];
    const float br = planeN[base + d.off_br];

    const float top = fmaf(tr - tl, d.lx, tl);
    const float bot = fmaf(br - bl, d.lx, bl);
    float v = fmaf(bot - top, d.ly, top);
    v *= d.valmul;

    __builtin_nontemporal_store(v, &outB[e]);   // streaming store, keep L2 for img
  }
}

} // namespace

extern "C" void kernel_launch(void* const* d_in, const int* in_sizes, int n_in,
                              void* d_out, int out_size, void* d_ws, size_t ws_size,
                              hipStream_t stream) {
  (void)in_sizes; (void)n_in; (void)d_ws; (void)ws_size; (void)out_size;
  const float* img  = (const float*)d_in[0];
  const float* rois = (const float*)d_in[1];
  // d_in[2] (input_image) only contributes its static 1024x1024 shape.
  float* out = (float*)d_out;

  dim3 grid(kN * kB);
  dim3 block(kThreads);
  roi_align_kernel<<<grid, block, 0, stream>>>(img, rois, out);
}